// LRUCombined_80161269613380
// MI455X (gfx1250) — compile-verified
//
#include <hip/hip_runtime.h>
#include <hip/hip_bf16.h>

// ---------------------------------------------------------------------------
// LRU forward for MI455X (gfx1250), fp32 WMMA path + async global->LDS DMA.
//   I = H = O = 1024, T = 8192
//   d_out layout: [ final_hidden interleaved re/im : 2048 floats ][ outputs O*T ]
//   d_ws  layout: [ aRe 1024 ][ aIm 1024 ][ gamma 1024 ][ pad 1024 ]
//                 [ bRe/hRe H*T ][ bIm/(-hIm) H*T ]
// ---------------------------------------------------------------------------

typedef __attribute__((ext_vector_type(2))) float v2f;
typedef __attribute__((ext_vector_type(8))) float v8f;

#define LRU_I 1024
#define LRU_H 1024
#define LRU_O 1024
#define LRU_T 8192

#define BM 128
#define BN 128
#define BK 32
#define AS_STRIDE (BK + 4)   // 36 floats -> 144B rows: 16B-aligned (async b128 ok)
#define XS_STRIDE (BK + 2)   // 34 floats -> 136B rows: 8B-aligned (b64 frags ok)

// ---------------------------------------------------------------------------
// gfx1250 async global->LDS copy (ASYNCcnt-tracked DMA, no VGPR round trip)
// ---------------------------------------------------------------------------
__device__ __forceinline__ unsigned lds_offset_of(const void* p) {
  // generic shared pointer: low 32 bits are the LDS byte offset
  return (unsigned)(uintptr_t)p;
}

__device__ __forceinline__ void async_copy16(unsigned lds_byte_addr,
                                             const float* sbase,
                                             unsigned voffset_bytes) {
  asm volatile("global_load_async_to_lds_b128 %0, %1, %2"
               :: "v"(lds_byte_addr), "v"(voffset_bytes), "s"(sbase)
               : "memory");
}

__device__ __forceinline__ void wait_async0() {
  asm volatile("s_wait_asynccnt 0x0" ::: "memory");
}

// ---------------------------------------------------------------------------
// Kernel 1: per-channel decay a = exp(-exp(nu) + i exp(theta)), gamma
// ---------------------------------------------------------------------------
__global__ void lru_params_kernel(const float* __restrict__ nu,
                                  const float* __restrict__ theta,
                                  float* __restrict__ aRe,
                                  float* __restrict__ aIm,
                                  float* __restrict__ gamma,
                                  int H) {
  int i = blockIdx.x * blockDim.x + threadIdx.x;
  if (i < H) {
    float en = __expf(nu[i]);
    float et = __expf(theta[i]);
    float r  = __expf(-en);                  // |a| = exp(-exp(nu))
    float s, c;
    __sincosf(et, &s, &c);
    aRe[i] = r * c;
    aIm[i] = r * s;
    gamma[i] = sqrtf(fmaxf(1.0f - r * r, 0.0f));
  }
}

// ---------------------------------------------------------------------------
// Kernel 2: fused multi-segment fp32 WMMA GEMM
//   Out[M,N] = rowScale[m] * sum_seg (Aseg[M,K] @ Xseg[K,N])
//   Block tile 128x128, 4 waves; wave tile 64x64 = 4x4 WMMA 16x16x4 tiles.
//   A tile: async-DMA'd into LDS (double buffered, row-major, K contiguous).
//   X tile: register-prefetched, stored transposed [BN][BK] so B fragments
//           are single contiguous ds_load_b64 (same K-half split as A).
// ---------------------------------------------------------------------------
__global__ __launch_bounds__(128, 1)
void wmma_gemm_f32_fused(const float* __restrict__ A0, const float* __restrict__ A1,
                         const float* __restrict__ A2,
                         const float* __restrict__ X0, const float* __restrict__ X1,
                         const float* __restrict__ X2,
                         int nseg,
                         float* __restrict__ Out,
                         const float* __restrict__ rowScale,   // may be null
                         int M, int N, int K) {
  __shared__ __align__(16) float As[2][BM * AS_STRIDE];   // 2 x 18 KB
  __shared__ __align__(16) float Xs[2][BN * XS_STRIDE];   // 2 x 17 KB

  const float* Aseg[3] = {A0, A1, A2};
  const float* Xseg[3] = {X0, X1, X2};

  const int tid   = threadIdx.x;
  const int lane  = tid & 31;
  const int wave  = tid >> 5;            // 0..3
  const int wrow  = (wave & 1) * 64;     // 2 row groups of 64
  const int wcol  = (wave >> 1) * 64;    // 2 col groups of 64

  const int blockM = blockIdx.y * BM;
  const int blockN = blockIdx.x * BN;

  const int lm    = lane & 15;           // M (A frag) / N (B frag, C tile)
  const int khalf = lane >> 4;           // K-half select per ISA layout

  const int kTiles = K / BK;
  const int nIters = nseg * kTiles;

  v8f acc[4][4];
#pragma unroll
  for (int i = 0; i < 4; ++i)
#pragma unroll
    for (int j = 0; j < 4; ++j)
      acc[i][j] = (v8f){0.f, 0.f, 0.f, 0.f, 0.f, 0.f, 0.f, 0.f};

  // ---- staging helpers -----------------------------------------------------
  // A tile 128x32: 1024 x 16B chunks, 8 per thread, via async DMA.
  auto stage_A_async = [&](int it, int buf) {
    const int seg = it / kTiles;
    const int k0  = (it - seg * kTiles) * BK;
    const float* sbase = Aseg[seg] + (size_t)blockM * K + k0;
    const unsigned ldsbase = lds_offset_of(&As[buf][0]);
#pragma unroll
    for (int p = 0; p < 8; ++p) {
      int c    = tid + 128 * p;
      int row  = c >> 3;            // 0..127
      int colf = (c & 7) * 4;       // 0..28
      async_copy16(ldsbase + (unsigned)(row * AS_STRIDE + colf) * 4,
                   sbase, (unsigned)(row * K + colf) * 4);
    }
  };

  // X tile 32x128: coalesced float4 global loads into registers.
  float4 xreg[8];
  auto load_X = [&](int it) {
    const int seg = it / kTiles;
    const int k0  = (it - seg * kTiles) * BK;
    const float* X = Xseg[seg];
    const int c4 = (tid & 31) * 4;
    const int r0 = tid >> 5;
#pragma unroll
    for (int p = 0; p < 8; ++p) {
      int r = r0 + 4 * p;
      xreg[p] = *(const float4*)&X[(size_t)(k0 + r) * N + blockN + c4];
    }
  };
  // transpose on store: Xs[n][k]
  auto store_X = [&](int buf) {
    float* xs = &Xs[buf][0];
    const int c4 = (tid & 31) * 4;
    const int r0 = tid >> 5;
#pragma unroll
    for (int p = 0; p < 8; ++p) {
      int r = r0 + 4 * p;
      xs[(c4 + 0) * XS_STRIDE + r] = xreg[p].x;
      xs[(c4 + 1) * XS_STRIDE + r] = xreg[p].y;
      xs[(c4 + 2) * XS_STRIDE + r] = xreg[p].z;
      xs[(c4 + 3) * XS_STRIDE + r] = xreg[p].w;
    }
  };

  // ---- pipeline ------------------------------------------------------------
  stage_A_async(0, 0);
  load_X(0);
  store_X(0);
  wait_async0();
  __syncthreads();

  for (int it = 0; it < nIters; ++it) {
    const int p = it & 1;
    const bool more = (it + 1) < nIters;

    if (more) {
      stage_A_async(it + 1, 1 - p);   // DMA overlaps the WMMA loop below
      load_X(it + 1);                 // global latency hidden behind compute
    }

    const float* as = &As[p][0];
    const float* xs = &Xs[p][0];
#pragma unroll
    for (int kk = 0; kk < BK; kk += 4) {
      v2f af[4], bf[4];
#pragma unroll
      for (int i = 0; i < 4; ++i)
        af[i] = *(const v2f*)&as[(wrow + i * 16 + lm) * AS_STRIDE + kk + 2 * khalf];
#pragma unroll
      for (int j = 0; j < 4; ++j)
        bf[j] = *(const v2f*)&xs[(wcol + j * 16 + lm) * XS_STRIDE + kk + 2 * khalf];
#pragma unroll
      for (int i = 0; i < 4; ++i)
#pragma unroll
        for (int j = 0; j < 4; ++j)
          acc[i][j] = __builtin_amdgcn_wmma_f32_16x16x4_f32(
              /*neg_a=*/false, af[i], /*neg_b=*/false, bf[j],
              /*c_mod=*/(short)0, acc[i][j],
              /*reuse_a=*/false, /*reuse_b=*/false);
    }

    if (more) {
      store_X(1 - p);   // write next X buffer (disjoint from buffer p)
      wait_async0();    // our async A chunks have landed
    }
    __syncthreads();    // everyone's chunks visible; reads of buf p done
  }

  // ---- epilogue: C/D 16x16 layout — lane: n = lm, VGPR r -> m = r + 8*khalf
#pragma unroll
  for (int i = 0; i < 4; ++i) {
#pragma unroll
    for (int r = 0; r < 8; ++r) {
      const int m = blockM + wrow + i * 16 + 8 * khalf + r;
      const float sc = rowScale ? rowScale[m] : 1.0f;
#pragma unroll
      for (int j = 0; j < 4; ++j) {
        const int n = blockN + wcol + j * 16 + lm;
        Out[(size_t)m * N + n] = acc[i][j][r] * sc;
      }
    }
  }
}

// ---------------------------------------------------------------------------
// Kernel 3: per-channel complex first-order scan (in place over b -> h).
//   One wave32 per channel; Hillis-Steele scan per 32-chunk + sequential carry.
//   Stores +h.re and NEGATED h.im into the planes (folds the "- C_imag @ h.im"
//   sign into the fused output GEMM); true h[:,T-1] goes to final_hidden.
// ---------------------------------------------------------------------------
__global__ __launch_bounds__(32, 8)
void lru_scan_kernel(float* __restrict__ hRe, float* __restrict__ hIm,
                     const float* __restrict__ aRe_, const float* __restrict__ aIm_,
                     float* __restrict__ finalHidden,  // 2*H floats interleaved
                     int T) {
  const int ch   = blockIdx.x;
  const int lane = threadIdx.x;
  const float ar = aRe_[ch];
  const float ai = aIm_[ch];

  float cr = 0.f, ci = 0.f;  // carry h_{-1} = 0
  float* rowR = hRe + (size_t)ch * T;
  float* rowI = hIm + (size_t)ch * T;

  for (int t0 = 0; t0 < T; t0 += 32) {
    float br = rowR[t0 + lane];
    float bi = rowI[t0 + lane];
    float Ar = ar, Ai = ai;

#pragma unroll
    for (int d = 1; d < 32; d <<= 1) {
      float pAr = __shfl_up(Ar, d);
      float pAi = __shfl_up(Ai, d);
      float pbr = __shfl_up(br, d);
      float pbi = __shfl_up(bi, d);
      if (lane >= d) {
        float nAr = pAr * Ar - pAi * Ai;
        float nAi = pAr * Ai + pAi * Ar;
        float nbr = Ar * pbr - Ai * pbi + br;   // cur.A * prev.b + cur.b
        float nbi = Ar * pbi + Ai * pbr + bi;
        Ar = nAr; Ai = nAi; br = nbr; bi = nbi;
      }
    }
    // h_k = A_k * carry + scan_b_k   (A_k = a^{k+1})
    float hr = Ar * cr - Ai * ci + br;
    float hi = Ar * ci + Ai * cr + bi;
    rowR[t0 + lane] = hr;
    rowI[t0 + lane] = -hi;          // negated plane for the fused output GEMM
    cr = __shfl(hr, 31);            // carry uses the TRUE h
    ci = __shfl(hi, 31);
  }
  if (lane == 0) {
    finalHidden[2 * ch + 0] = cr;
    finalHidden[2 * ch + 1] = ci;
  }
}

// ---------------------------------------------------------------------------
// Launcher
// ---------------------------------------------------------------------------
extern "C" void kernel_launch(void* const* d_in, const int* in_sizes, int n_in,
                              void* d_out, int out_size, void* d_ws, size_t ws_size,
                              hipStream_t stream) {
  (void)in_sizes; (void)n_in; (void)out_size; (void)ws_size;

  const float* x      = (const float*)d_in[0];   // (I, T)
  const float* B_real = (const float*)d_in[1];   // (H, I)
  const float* B_imag = (const float*)d_in[2];   // (H, I)
  const float* nu     = (const float*)d_in[3];   // (H,)
  const float* theta  = (const float*)d_in[4];   // (H,)
  const float* C_real = (const float*)d_in[5];   // (O, H)
  const float* C_imag = (const float*)d_in[6];   // (O, H)
  const float* Dm     = (const float*)d_in[7];   // (O, I)

  float* out = (float*)d_out;
  float* ws  = (float*)d_ws;

  float* aRe   = ws;
  float* aIm   = ws + 1024;
  float* gamma = ws + 2048;
  float* bRe   = ws + 4096;                        // H*T floats (becomes hRe)
  float* bIm   = bRe + (size_t)LRU_H * LRU_T;      // H*T floats (becomes -hIm)

  // 1) per-channel parameters
  lru_params_kernel<<<4, 256, 0, stream>>>(nu, theta, aRe, aIm, gamma, LRU_H);

  // 2) b = gamma * (B @ x)   (real & imag planes)
  dim3 grid(LRU_T / BN, LRU_H / BM);   // (64, 8)
  wmma_gemm_f32_fused<<<grid, 128, 0, stream>>>(
      B_real, nullptr, nullptr, x, nullptr, nullptr, /*nseg=*/1,
      bRe, gamma, LRU_H, LRU_T, LRU_I);
  wmma_gemm_f32_fused<<<grid, 128, 0, stream>>>(
      B_imag, nullptr, nullptr, x, nullptr, nullptr, /*nseg=*/1,
      bIm, gamma, LRU_H, LRU_T, LRU_I);

  // 3) recurrence scan, in place; writes final_hidden to d_out[0:2048]
  lru_scan_kernel<<<LRU_H, 32, 0, stream>>>(bRe, bIm, aRe, aIm, out, LRU_T);

  // 4) outputs = C_real@h.re + C_imag@(-h.im) + D@x  (single fused pass)
  float* outputs = out + 2 * LRU_H;
  wmma_gemm_f32_fused<<<grid, 128, 0, stream>>>(
      C_real, C_imag, Dm, bRe, bIm, x, /*nseg=*/3,
      outputs, nullptr, LRU_O, LRU_T, LRU_H);
}